// FACoef_35132832481991
// MI455X (gfx1250) — compile-verified
//
#include <hip/hip_runtime.h>
#include <cstdint>

// FACoef on MI455X (gfx1250):
//   s_k(b) = sum of entries of x_b^k, k=2..5, via matvecs only:
//     v1 = X*1, u1 = X^T*1, v2 = X*v1, u2 = X^T*u1, w = X*v2
//     s2=u1.v1  s3=u1.v2  s4=u2.v2  s5=u2.w
//   Memory-bound: one pass of x over HBM (134 MB -> ~5.8us @ 23.3 TB/s).
//   X tile lives in LDS (async-to-LDS path), matvecs use v_wmma_f32_16x16x4_f32.

typedef __attribute__((ext_vector_type(2))) float v2f;
typedef __attribute__((ext_vector_type(8))) float v8f;

#define NDIM 128
#define LDSTRIDE 132   // 132 % 64 = 4 -> conflict-free strided rows; 528B rows stay 16B aligned

// y[r0..r0+15] = sum_j X[r0+m, j] * v[j]        (v == nullptr means v = ones)
// A(16x4) = X tile rows, B(4x16) = v chunk broadcast to all 16 columns,
// so every column of D holds the matvec partial; accumulate over 32 K-chunks.
__device__ __forceinline__ void wave_matvec_Xv(const float* Xs, const float* v, float* y,
                                               int wave, int lane) {
  const int r0   = wave * 16;
  const int half = lane >> 4;   // lanes 16-31 carry K = k0+2, k0+3 per ISA A-layout
  const int m    = lane & 15;
  v8f acc = {};
  #pragma unroll
  for (int k0 = 0; k0 < NDIM; k0 += 4) {
    const int k = k0 + 2 * half;
    v2f a, b;
    a.x = Xs[(r0 + m) * LDSTRIDE + k];
    a.y = Xs[(r0 + m) * LDSTRIDE + k + 1];
    b.x = v ? v[k]     : 1.0f;
    b.y = v ? v[k + 1] : 1.0f;
    acc = __builtin_amdgcn_wmma_f32_16x16x4_f32(false, a, false, b, (short)0, acc,
                                                false, false);
  }
  // D layout: VGPR r, lanes 0-15 -> M=r ; lanes 16-31 -> M=r+8. Columns identical.
  if (m == 0) {
    #pragma unroll
    for (int r = 0; r < 8; ++r) y[r0 + half * 8 + r] = acc[r];
  }
}

// y[c0..c0+15] = sum_i u[i] * X[i, c0+n]        (u == nullptr means u = ones)
// A(16x4) rows all equal to the u chunk, B(4x16) = X rows k0..k0+3 striped across lanes,
// so every row of D holds the transposed-matvec partial.
__device__ __forceinline__ void wave_matvec_XTu(const float* Xs, const float* u, float* y,
                                                int wave, int lane) {
  const int c0   = wave * 16;
  const int half = lane >> 4;
  const int m    = lane & 15;
  v8f acc = {};
  #pragma unroll
  for (int k0 = 0; k0 < NDIM; k0 += 4) {
    const int k = k0 + 2 * half;
    v2f a, b;
    a.x = u ? u[k]     : 1.0f;   // A[m,k] broadcast over m
    a.y = u ? u[k + 1] : 1.0f;
    b.x = Xs[k * LDSTRIDE + c0 + m];        // B[k, n]
    b.y = Xs[(k + 1) * LDSTRIDE + c0 + m];  // B[k+1, n]
    acc = __builtin_amdgcn_wmma_f32_16x16x4_f32(false, a, false, b, (short)0, acc,
                                                false, false);
  }
  if (half == 0) y[c0 + m] = acc[0];   // row M=0, N=lane
}

__global__ __launch_bounds__(256) void facoef_kernel(const float* __restrict__ x,
                                                     const float* __restrict__ coef,
                                                     float* __restrict__ out) {
  extern __shared__ float smem[];
  float* Xs = smem;                        // 128 x 132
  float* v1 = smem + NDIM * LDSTRIDE;
  float* u1 = v1 + NDIM;
  float* v2 = u1 + NDIM;
  float* u2 = v2 + NDIM;
  float* w3 = u2 + NDIM;
  float* sd = w3 + NDIM;                   // 4 dot results

  const int tid  = threadIdx.x;
  const int lane = tid & 31;
  const int wave = tid >> 5;
  const int b    = blockIdx.x;
  const float* xb = x + (size_t)b * (NDIM * NDIM);

  // ---- async copy of the 64KB sample into LDS (gfx1250 async-tensor path) ----
  // low 32 bits of a flat shared pointer are the LDS offset.
  const unsigned ldsXs = (unsigned)(uintptr_t)Xs;
  for (int r = wave; r < NDIM; r += 8) {
    unsigned lds_addr = ldsXs + (unsigned)((r * LDSTRIDE + lane * 4) * 4);
    const float* src  = xb + r * NDIM + lane * 4;   // 16B per lane, one row per wave-op
    asm volatile("global_load_async_to_lds_b128 %0, %1, off"
                 :: "v"(lds_addr), "v"(src) : "memory");
  }
#if __has_builtin(__builtin_amdgcn_s_wait_asynccnt)
  __builtin_amdgcn_s_wait_asynccnt(0);
#else
  asm volatile("s_wait_asynccnt 0" ::: "memory");
#endif
  __syncthreads();

  // ---- five matvecs, WMMA-powered, barriers between dependent passes ----
  wave_matvec_Xv (Xs, nullptr, v1, wave, lane);   // v1 = X * 1
  wave_matvec_XTu(Xs, nullptr, u1, wave, lane);   // u1 = X^T * 1
  __syncthreads();
  wave_matvec_Xv (Xs, v1, v2, wave, lane);        // v2 = X * v1
  wave_matvec_XTu(Xs, u1, u2, wave, lane);        // u2 = X^T * u1
  __syncthreads();
  wave_matvec_Xv (Xs, v2, w3, wave, lane);        // w  = X * v2
  __syncthreads();

  // ---- dots: s2=u1.v1  s3=u1.v2  s4=u2.v2  s5=u2.w  (one wave each) ----
  if (wave < 4) {
    const float* p = (wave < 2) ? u1 : u2;
    const float* q = (wave == 0) ? v1 : (wave == 1 || wave == 2) ? v2 : w3;
    float partial = 0.f;
    for (int i = lane; i < NDIM; i += 32) partial += p[i] * q[i];
    #pragma unroll
    for (int off = 16; off > 0; off >>= 1) partial += __shfl_xor(partial, off, 32);
    if (lane == 0) sd[wave] = partial;
  }
  __syncthreads();

  // ---- polynomial combine: res = sum_ij coef[i,j] * S_i^(j+1) / n^(i+j+2) ----
  if (tid == 0) {
    const float n = (float)(NDIM * NDIM);
    float res = 0.f;
    float npow_i = n * n;                 // n^(i+2) at i=0
    #pragma unroll
    for (int i = 0; i < 4; ++i) {
      const float Si = sd[i];
      float sp = 1.0f;
      float dn = npow_i;                  // n^(i+j+2) at j=0
      #pragma unroll
      for (int j = 0; j < 4; ++j) {
        sp *= Si;                         // S_i^(j+1)
        res += coef[i * 4 + j] * sp / dn; // n^9 = 8.4e37 still finite in fp32
        dn *= n;
      }
      npow_i *= n;
    }
    out[b] = res;
  }
}

extern "C" void kernel_launch(void* const* d_in, const int* in_sizes, int n_in,
                              void* d_out, int out_size, void* d_ws, size_t ws_size,
                              hipStream_t stream) {
  const float* x    = (const float*)d_in[0];
  const float* coef = (const float*)d_in[1];
  float* out        = (float*)d_out;
  const int B = out_size;                 // 2048 samples, one workgroup each

  const size_t smem = (size_t)(NDIM * LDSTRIDE + 5 * NDIM + 8) * sizeof(float); // ~70 KB < 320 KB/WGP
  facoef_kernel<<<dim3(B), dim3(256), smem, stream>>>(x, coef, out);
}